// KGather_4088808866303
// MI455X (gfx1250) — compile-verified
//
#include <hip/hip_runtime.h>
#include <stdint.h>

// Problem constants (from reference): N=16, P2=49, TOPK=8, W2=64, CK=128
#define N_       16
#define P2_      49
#define TOPK_    8
#define W2_      64
#define CK_      128
#define ROWF     (W2_ * CK_)        // 8192 floats per (n,p,t) tile = 32 KB
#define NTHREADS 256
#define CHUNKF   (NTHREADS * 4)     // 1024 floats per chunk (one float4 per lane)
#define NCHUNK   (ROWF / CHUNKF)    // 8 chunks per row

typedef float f32x4 __attribute__((ext_vector_type(4)));

#if defined(__has_builtin)
#  if __has_builtin(__builtin_amdgcn_s_wait_asynccnt)
#    define HAVE_WAIT_ASYNC 1
#  endif
#endif

#if defined(HAVE_WAIT_ASYNC)
#  define WAIT_ASYNC(n) __builtin_amdgcn_s_wait_asynccnt(n)
#else
#  define WAIT_ASYNC(n) asm volatile("s_wait_asynccnt " #n ::: "memory")
#endif

typedef __attribute__((address_space(3))) float lfloat;   // LDS float
typedef const __attribute__((address_space(3))) f32x4 lcf32x4;

// gfx1250 async copy: global (16B per lane) -> LDS, tracked by ASYNCcnt.
// VDST = per-lane LDS byte address, VADDR = per-lane 64-bit global address.
__device__ __forceinline__ void async_copy_b128(const float* gsrc,
                                                unsigned lds_byte_addr) {
  asm volatile("global_load_async_to_lds_b128 %0, %1, off"
               :
               : "v"(lds_byte_addr), "v"(gsrc)
               : "memory");
}

// One block per output row (n,p,t). Gather source row, scale by routing
// weight, stream out with non-temporal 128-bit stores. Source chunks are
// staged via the gfx1250 async global->LDS path with a 2-deep pipeline.
__global__ __launch_bounds__(NTHREADS) void kgather_scale(
    const int* __restrict__ r_idx,
    const float* __restrict__ r_weight,
    const float* __restrict__ k,
    float* __restrict__ out)
{
  const int row = blockIdx.x;                      // (n*P2 + p)*TOPK + t
  const int n   = row / (P2_ * TOPK_);
  const int idx = r_idx[row];                      // block-uniform -> s_load
  const float w = r_weight[row];                   // block-uniform -> s_load

  const float* __restrict__ src = k + (size_t)(n * P2_ + idx) * ROWF;
  float* __restrict__ dst = out + (size_t)row * ROWF;
  const int off = threadIdx.x * 4;                 // lane's float4 slot in a chunk

  __shared__ float lds[2 * CHUNKF];                // 8 KB double buffer
  lfloat* lp0 = (lfloat*)&lds[off];                // per-lane private 16B slot
  lfloat* lp1 = (lfloat*)&lds[CHUNKF + off];
  const unsigned l0 = (unsigned)(uintptr_t)lp0;    // LDS byte offsets for asm
  const unsigned l1 = (unsigned)(uintptr_t)lp1;

  // Prime the pipeline with chunk 0.
  async_copy_b128(src + off, l0);

#pragma unroll
  for (int i = 0; i < NCHUNK; ++i) {
    if (i + 1 < NCHUNK) {
      async_copy_b128(src + (i + 1) * CHUNKF + off, ((i + 1) & 1) ? l1 : l0);
      WAIT_ASYNC(1);                               // chunk i has landed in LDS
    } else {
      WAIT_ASYNC(0);
    }
    f32x4 v = *(lcf32x4*)((i & 1) ? lp1 : lp0);    // ds_load_b128, own slot only
    v *= w;
    __builtin_nontemporal_store(v, (f32x4*)(dst + i * CHUNKF + off));
  }
}

extern "C" void kernel_launch(void* const* d_in, const int* in_sizes, int n_in,
                              void* d_out, int out_size, void* d_ws, size_t ws_size,
                              hipStream_t stream) {
  (void)in_sizes; (void)n_in; (void)out_size; (void)d_ws; (void)ws_size;
  const int*   r_idx    = (const int*)d_in[0];     // (N,P2,TOPK) int
  const float* r_weight = (const float*)d_in[1];   // (N,P2,TOPK) f32
  const float* k        = (const float*)d_in[2];   // (N,P2,W2,CK) f32
  float* out = (float*)d_out;                      // (N,P2,TOPK,W2,CK) f32

  dim3 grid(N_ * P2_ * TOPK_);                     // 6272 blocks, one per row
  kgather_scale<<<grid, NTHREADS, 0, stream>>>(r_idx, r_weight, k, out);
}